// DTNN_layer_74680891342857
// MI455X (gfx1250) — compile-verified
//
#include <hip/hip_runtime.h>

typedef __attribute__((ext_vector_type(2))) float v2f;
typedef __attribute__((ext_vector_type(4))) float v4f;
typedef __attribute__((ext_vector_type(8))) float v8f;

#define BN 1024   // B*N rows
#define D  128    // D_IN == D_DIST == H == D_OUT
#define NJ 256    // j extent

// ---------------------------------------------------------------------------
// Kernel 1: Dsum[r, d] = sum_j distance[r, j, d]      (r = b*N + i)
// Pure HBM stream of 134 MB -> the whole cost of this layer (~5.8 us roofline).
// One block per r; wave w handles j = w, w+8, ...; lanes sweep d in float4.
// Non-temporal loads: data is single-use, keep it out of L2 retention.
// ---------------------------------------------------------------------------
__global__ __launch_bounds__(256) void reduce_j_kernel(
    const float* __restrict__ distance, float* __restrict__ dsum)
{
    const int r    = blockIdx.x;
    const int lane = threadIdx.x & 31;   // d-group: d = lane*4
    const int wave = threadIdx.x >> 5;   // j-stream: 0..7
    const int d    = lane << 2;

    const float* base = distance + (size_t)r * NJ * D + d;

    v4f acc = {0.f, 0.f, 0.f, 0.f};
    #pragma unroll 4
    for (int j = wave; j < NJ; j += 8) {
        v4f v = __builtin_nontemporal_load((const v4f*)(base + (size_t)j * D));
        acc += v;
    }

    __shared__ v4f part[8][32];
    part[wave][lane] = acc;
    __syncthreads();

    if (wave == 0) {
        #pragma unroll
        for (int g = 1; g < 8; ++g) acc += part[g][lane];
        *(v4f*)(dsum + (size_t)r * D + d) = acc;
    }
}

// ---------------------------------------------------------------------------
// Kernel 2: fused  hx = x@Wcf^T + bcf
//                  S  = Dsum@Wdf^T + 256*bdf
//                  z2 = hx * S
//                  out = z2@Wfc^T
// 64 blocks (16 rows each) x 8 waves; wave w owns output columns [16w,16w+16).
// V_WMMA_F32_16X16X4_F32, K reduced in steps of 4 (uniform loop bound; the
// lane-half K offset khi is applied inside the body so EXEC stays all-1s).
// A-frag layout (16x4 f32): lanes 0-15 hold M=lane, K=(k0,k0+1);
//                           lanes 16-31 hold M=lane-16, K=(k0+2,k0+3).
// B-frag mirrored with N on the lane axis. C/D: VGPR v = row M=v (lo half) /
// M=v+8 (hi half), N = lane&15.
// ---------------------------------------------------------------------------
__global__ __launch_bounds__(256) void fused_gemm_kernel(
    const float* __restrict__ x,
    const float* __restrict__ dsum,
    const float* __restrict__ Wcf_w, const float* __restrict__ Wcf_b,
    const float* __restrict__ Wdf_w, const float* __restrict__ Wdf_b,
    const float* __restrict__ Wfc_w,
    float* __restrict__ out)
{
    __shared__ float z2s[16 * 132];          // padded rows: conflict-free

    const int row0 = blockIdx.x << 4;        // 16 rows per block
    const int lane = threadIdx.x & 31;
    const int wave = threadIdx.x >> 5;
    const int n0   = wave << 4;              // this wave's output column tile
    const int lm   = lane & 15;              // M (for A) / N (for B,C,D)
    const int khi  = (lane >> 4) << 1;       // 0 (lanes 0-15) or 2 (lanes 16-31)
    const int mb   = (lane >> 4) << 3;       // C/D row base: 0 or 8

    const float* xrow = x     + (size_t)(row0 + lm) * D;
    const float* drow = dsum  + (size_t)(row0 + lm) * D;
    const float* wcf  = Wcf_w + (size_t)(n0 + lm) * D;
    const float* wdf  = Wdf_w + (size_t)(n0 + lm) * D;
    const float* wfc  = Wfc_w + (size_t)(n0 + lm) * D;

    v8f acc_hx = {};
    v8f acc_s  = {};
    for (int k0 = 0; k0 < D; k0 += 4) {      // uniform bound: EXEC stays full
        const int k = k0 + khi;
        v2f ax = *(const v2f*)(xrow + k);
        v2f bx = *(const v2f*)(wcf  + k);
        v2f ad = *(const v2f*)(drow + k);
        v2f bd = *(const v2f*)(wdf  + k);
        acc_hx = __builtin_amdgcn_wmma_f32_16x16x4_f32(
            false, ax, false, bx, (short)0, acc_hx, false, false);
        acc_s  = __builtin_amdgcn_wmma_f32_16x16x4_f32(
            false, ad, false, bd, (short)0, acc_s,  false, false);
    }

    // biases (bias of hd summed over j contributes N=256 copies), elementwise
    const float bcf = Wcf_b[n0 + lm];
    const float bdf = Wdf_b[n0 + lm] * 256.0f;
    #pragma unroll
    for (int v = 0; v < 8; ++v) {
        float hx = acc_hx[v] + bcf;
        float s  = acc_s[v]  + bdf;
        z2s[(mb + v) * 132 + n0 + lm] = hx * s;
    }
    __syncthreads();

    v8f acc_o = {};
    for (int k0 = 0; k0 < D; k0 += 4) {
        const int k = k0 + khi;
        v2f a = *(const v2f*)(&z2s[lm * 132 + k]);
        v2f b = *(const v2f*)(wfc + k);
        acc_o = __builtin_amdgcn_wmma_f32_16x16x4_f32(
            false, a, false, b, (short)0, acc_o, false, false);
    }

    #pragma unroll
    for (int v = 0; v < 8; ++v)
        out[(size_t)(row0 + mb + v) * D + n0 + lm] = acc_o[v];
}

// ---------------------------------------------------------------------------
extern "C" void kernel_launch(void* const* d_in, const int* in_sizes, int n_in,
                              void* d_out, int out_size, void* d_ws, size_t ws_size,
                              hipStream_t stream)
{
    const float* x     = (const float*)d_in[0];
    const float* dist  = (const float*)d_in[1];
    const float* Wcf_w = (const float*)d_in[2];
    const float* Wcf_b = (const float*)d_in[3];
    const float* Wdf_w = (const float*)d_in[4];
    const float* Wdf_b = (const float*)d_in[5];
    const float* Wfc_w = (const float*)d_in[6];
    float* out  = (float*)d_out;
    float* dsum = (float*)d_ws;              // 1024*128 floats = 512 KB scratch

    reduce_j_kernel<<<dim3(BN), dim3(256), 0, stream>>>(dist, dsum);
    fused_gemm_kernel<<<dim3(BN / 16), dim3(256), 0, stream>>>(
        x, dsum, Wcf_w, Wcf_b, Wdf_w, Wdf_b, Wfc_w, out);
}